// AliasFreeActivation_5669356832303
// MI455X (gfx1250) — compile-verified
//
#include <hip/hip_runtime.h>
#include <hip/hip_bf16.h>

typedef float v2f __attribute__((ext_vector_type(2)));
typedef float v8f __attribute__((ext_vector_type(8)));

#define WMMA_F32(a, b, c) \
  __builtin_amdgcn_wmma_f32_16x16x4_f32(false, (a), false, (b), (short)0, (c), false, false)

namespace {
constexpr int HIN   = 84;    // size_p + 2*MARGIN = 64 + 20
constexpr int WIN   = 84;
constexpr int HOUT  = 148;   // 168 - 2*10 crop
constexpr int WOUT  = 148;
constexpr float SLOPE = 0.2f;
}

// Fully fused: up(x4) H, up(x4) V, leaky-relu, down(/2) H, down(/2) V, crop.
// One wave (32 threads) per 16x16 output tile; all four FIR passes are WMMA
// matmuls against tile-invariant banded coefficient matrices in LDS.
// K-loops are trimmed to the nonzero bands:
//   stage 1/2: band m in [4t, 4t+9]  -> 3 K-steps of 4 per 16-wide tile
//   stage 3/4: band q in [0, 41]     -> 11 K-steps of 4
__global__ __launch_bounds__(32)
void alias_free_act_fused(const float* __restrict__ x,
                          const float* __restrict__ kup,   // 24 taps, up factor 4
                          const float* __restrict__ kdn,   // 12 taps, down factor 2
                          float* __restrict__ out)
{
  __shared__ float kuS[24];
  __shared__ float kdS[12];
  __shared__ float Xs[32][21];   // 18x18 valid input tile, zero padded
  __shared__ float Wu[20][48];   // up-FIR banded matrix: input-idx m -> up-idx q
  __shared__ float Wd[16][49];   // down-FIR banded matrix: out-idx r <- up-idx q
  __shared__ float T1[32][48];   // after horizontal upsample (input rows x up cols)
  __shared__ float Us[48][49];   // after vertical upsample + lrelu (up rows x up cols)
  __shared__ float Vs[48][16];   // after horizontal downsample (up rows x out cols)

  const int l  = threadIdx.x;
  const int lo = l & 15;         // N / M lane index inside fragment
  const int hi = l >> 4;         // half-wave select

  const int tx = blockIdx.x;     // output tile col, 0..9
  const int ty = blockIdx.y;     // output tile row, 0..9
  const int bc = blockIdx.z;     // flattened batch*channel, 0..511

  if (l < 24) kuS[l] = kup[l];
  if (l < 12) kdS[l] = kdn[l];
  __syncthreads();

  // ---- build tile-invariant banded coefficient matrices ----
  // Up (polyphase, factor 4): U[u0+q] = sum_j ku[4j + p] * X[i0 + m],
  //   p = (q+1)&3,  j = ((q+1)>>2) + 5 - m  (u0 = 32*tile+15, i0 = 8*tile+1)
  for (int idx = l; idx < 20 * 48; idx += 32) {
    const int m = idx / 48, q = idx % 48;
    const int p = (q + 1) & 3;
    const int j = ((q + 1) >> 2) + 5 - m;
    Wu[m][q] = (j >= 0 && j < 6) ? kuS[4 * j + p] : 0.0f;
  }
  // Down (stride 2): D[oy0+r] = sum_s kd[s] * U[u0 + 2r + 11 - s]
  for (int idx = l; idx < 16 * 48; idx += 32) {
    const int r = idx / 48, q = idx % 48;
    const int s = 2 * r + 11 - q;
    Wd[r][q] = (s >= 0 && s < 12) ? kdS[s] : 0.0f;
  }

  // ---- load input tile (18x18 valid; origin 8*tile+1; OOB -> 0) ----
  const int i0y = 8 * ty + 1;
  const int i0x = 8 * tx + 1;
  const float* __restrict__ xb = x + (size_t)bc * (HIN * WIN);
  for (int idx = l; idx < 32 * 21; idx += 32) {
    const int rr = idx / 21, cc = idx % 21;
    const int gy = i0y + rr, gx = i0x + cc;
    float v = 0.0f;
    if (rr < 18 && cc < 18 && gy < HIN && gx < WIN)
      v = xb[gy * WIN + gx];
    Xs[rr][cc] = v;
  }
  __syncthreads();

  // === stage 1: horizontal upsample   T1[32x48] = Xs(32x20) x Wu(20x48) ===
  // Band: for up-col tile nt, nonzero input cols m in [4nt, 4nt+9] -> 3 K-steps.
  for (int mt = 0; mt < 2; ++mt) {
    for (int nt = 0; nt < 3; ++nt) {
      v8f acc = {};
#pragma unroll
      for (int kk = 0; kk < 3; ++kk) {
        const int k0 = 4 * nt + 4 * kk;
        v2f a, b;
        a.x = Xs[mt * 16 + lo][k0 + 2 * hi + 0];
        a.y = Xs[mt * 16 + lo][k0 + 2 * hi + 1];
        b.x = Wu[k0 + 2 * hi + 0][nt * 16 + lo];
        b.y = Wu[k0 + 2 * hi + 1][nt * 16 + lo];
        acc = WMMA_F32(a, b, acc);
      }
#pragma unroll
      for (int v = 0; v < 8; ++v)
        T1[mt * 16 + v + 8 * hi][nt * 16 + lo] = acc[v];
    }
  }
  __syncthreads();

  // === stage 2: vertical upsample + leaky relu
  //     Us[48x48] = Wu^T(48x20) x T1(20x48) ===
  // Band: for up-row tile mt, nonzero input rows iy in [4mt, 4mt+9] -> 3 K-steps.
  for (int mt = 0; mt < 3; ++mt) {
    for (int nt = 0; nt < 3; ++nt) {
      v8f acc = {};
#pragma unroll
      for (int kk = 0; kk < 3; ++kk) {
        const int k0 = 4 * mt + 4 * kk;
        v2f a, b;
        a.x = Wu[k0 + 2 * hi + 0][mt * 16 + lo];   // A[m=q][k=iy] = Wu[iy][q]
        a.y = Wu[k0 + 2 * hi + 1][mt * 16 + lo];
        b.x = T1[k0 + 2 * hi + 0][nt * 16 + lo];
        b.y = T1[k0 + 2 * hi + 1][nt * 16 + lo];
        acc = WMMA_F32(a, b, acc);
      }
#pragma unroll
      for (int v = 0; v < 8; ++v) {
        float u = acc[v];
        u = (u > 0.0f) ? u : SLOPE * u;            // fused leaky relu
        Us[mt * 16 + v + 8 * hi][nt * 16 + lo] = u;
      }
    }
  }
  __syncthreads();

  // === stage 3: horizontal downsample  Vs[48x16] = Us(48x48) x Wd^T(48x16) ===
  // Band: Wd[r][q] == 0 for q >= 42 -> 11 K-steps (q in [0,43]).
  for (int mt = 0; mt < 3; ++mt) {
    v8f acc = {};
#pragma unroll
    for (int k0 = 0; k0 < 44; k0 += 4) {
      v2f a, b;
      a.x = Us[mt * 16 + lo][k0 + 2 * hi + 0];
      a.y = Us[mt * 16 + lo][k0 + 2 * hi + 1];
      b.x = Wd[lo][k0 + 2 * hi + 0];               // B[k=q][n=r] = Wd[r][q]
      b.y = Wd[lo][k0 + 2 * hi + 1];
      acc = WMMA_F32(a, b, acc);
    }
#pragma unroll
    for (int v = 0; v < 8; ++v)
      Vs[mt * 16 + v + 8 * hi][lo] = acc[v];
  }
  __syncthreads();

  // === stage 4: vertical downsample   O[16x16] = Wd(16x48) x Vs(48x16) ===
  {
    v8f acc = {};
#pragma unroll
    for (int k0 = 0; k0 < 44; k0 += 4) {
      v2f a, b;
      a.x = Wd[lo][k0 + 2 * hi + 0];               // A[m=r][k=q] = Wd[r][q]
      a.y = Wd[lo][k0 + 2 * hi + 1];
      b.x = Vs[k0 + 2 * hi + 0][lo];
      b.y = Vs[k0 + 2 * hi + 1][lo];
      acc = WMMA_F32(a, b, acc);
    }
    float* __restrict__ ob = out + (size_t)bc * (HOUT * WOUT);
    const int oy0 = 16 * ty, ox0 = 16 * tx;
#pragma unroll
    for (int v = 0; v < 8; ++v) {
      const int oy = oy0 + v + 8 * hi;
      const int ox = ox0 + lo;
      if (oy < HOUT && ox < WOUT)
        ob[oy * WOUT + ox] = acc[v];
    }
  }
}

extern "C" void kernel_launch(void* const* d_in, const int* in_sizes, int n_in,
                              void* d_out, int out_size, void* d_ws, size_t ws_size,
                              hipStream_t stream) {
  (void)in_sizes; (void)n_in; (void)out_size; (void)d_ws; (void)ws_size;
  const float* x  = (const float*)d_in[0];   // (2,256,84,84)
  const float* ku = (const float*)d_in[1];   // 24 taps
  const float* kd = (const float*)d_in[2];   // 12 taps
  float* out = (float*)d_out;                // (2,256,148,148)

  // 10x10 tiles of 16x16 cover 148x148; z = batch*channel = 2*256
  dim3 grid(10, 10, 2 * 256);
  dim3 block(32, 1, 1);
  alias_free_act_fused<<<grid, block, 0, stream>>>(x, ku, kd, out);
}